// SinkhornLog_90829968375846
// MI455X (gfx1250) — compile-verified
//
#include <hip/hip_runtime.h>
#include <hip/hip_bf16.h>

// Sinkhorn-Knopp (log space) on [256, 512, 512] f32, TAU=0.05, 10 iterations.
//
// MI455X strategy:
//  * Bandwidth-bound workload. Exploit the 192MB global L2: process batches in
//    groups of 64 (64MB working set, L2-resident) and run all 21 passes for a
//    group before moving on. HBM traffic ~= one read + one write of 256MB.
//  * Row-wise sum-of-exp is computed on the matrix pipe with
//    V_WMMA_F32_16X16X4_F32 against a ones B-matrix (row-sum as GEMV),
//    co-executing with v_exp_f32 TRANS ops on the VALU.
//  * Z tiles staged into LDS with GLOBAL_LOAD_ASYNC_TO_LDS_B128 (ASYNCcnt
//    path, no VGPR round-trip), padded stride 516 floats: bank-conflict-free
//    for the 16-row-strided WMMA A-layout reads, 16B-aligned for b128 copies.

#define N        512
#define BATCH    256
#define INV_TAU  20.0f
#define ITERS    10
#define GROUP_B  64
#define NGROUPS  (BATCH / GROUP_B)
#define ZS_STRIDE 516          // 512 + 4 pad floats per row (byte stride 2064 = 129*16)

typedef __attribute__((ext_vector_type(2))) float v2f;
typedef __attribute__((ext_vector_type(8))) float v8f;
typedef __attribute__((ext_vector_type(4))) int   v4i;

typedef __attribute__((address_space(1))) v4i gv4i;   // global int4
typedef __attribute__((address_space(3))) v4i sv4i;   // LDS int4

#define HAS_WMMA_F32X4 __has_builtin(__builtin_amdgcn_wmma_f32_16x16x4_f32)

#if defined(__AMDGCN__) && \
    __has_builtin(__builtin_amdgcn_global_load_async_to_lds_b128) && \
    __has_builtin(__builtin_amdgcn_s_wait_asynccnt)
#define USE_ASYNC_LDS 1
#else
#define USE_ASYNC_LDS 0
#endif

// ---------------------------------------------------------------------------
// u[b, r] = -logsumexp_c( logits[b,r,c]*INV_TAU + v[b,c] )
// One block = 128 rows of one batch; one wave = 16 rows (WMMA M dimension).
// ---------------------------------------------------------------------------
__global__ __launch_bounds__(256)
void sk_row_kernel(const float* __restrict__ logits, const float* __restrict__ v,
                   float* __restrict__ u, int b0)
{
    extern __shared__ float smem[];
    float* zs = smem;                       // 128 * ZS_STRIDE floats
    float* vs = smem + 128 * ZS_STRIDE;     // 512 floats

    const int t    = threadIdx.x;
    const int b    = b0 + (blockIdx.x >> 2);
    const int row0 = (blockIdx.x & 3) * 128;

    // stage v[b, :]
    vs[t]       = v[(size_t)b * N + t];
    vs[t + 256] = v[(size_t)b * N + t + 256];

    // stage Z tile (128 x 512 f32) with padded row stride
    const float4* zg = (const float4*)(logits + ((size_t)b * N + row0) * N);
#if USE_ASYNC_LDS
    {
        // CDNA5 async global->LDS copy: no VGPR return data, tracked by ASYNCcnt.
        #pragma unroll 4
        for (int i = t; i < 128 * 128; i += 256) {
            int rr = i >> 7;           // local row
            int c4 = i & 127;          // float4 column
            __builtin_amdgcn_global_load_async_to_lds_b128(
                (gv4i*)(zg + (size_t)rr * 128 + c4),
                (sv4i*)(zs + rr * ZS_STRIDE + (c4 << 2)),
                0, 0);
        }
        __builtin_amdgcn_s_wait_asynccnt(0);   // this wave's copies landed in LDS
    }
#else
    #pragma unroll 4
    for (int i = t; i < 128 * 128; i += 256) {
        int rr = i >> 7;           // local row
        int c4 = i & 127;          // float4 column
        float4 val = zg[(size_t)rr * 128 + c4];
        *(float4*)(zs + rr * ZS_STRIDE + (c4 << 2)) = val;
    }
#endif
    __syncthreads();

    const int lane = t & 31;
    const int wv   = t >> 5;                    // wave id: rows row0+16*wv ..+15
    const int lr   = (wv << 4) + (lane & 15);   // local row this lane feeds
    const int g    = lane >> 4;                 // K-pair group (A-layout)

    const float2* zrow = (const float2*)(zs + lr * ZS_STRIDE);

    // ---- pass 1: row max of (z*INV_TAU + v) -------------------------------
    float mx0 = -3.402823e38f, mx1 = -3.402823e38f;
    for (int j = 0; j < 128; j += 2) {
        float2 za = zrow[2 * j + g];
        float2 zb = zrow[2 * (j + 1) + g];
        int ca = 4 * j + 2 * g;
        int cb = ca + 4;
        mx0 = fmaxf(mx0, fmaxf(za.x * INV_TAU + vs[ca],     za.y * INV_TAU + vs[ca + 1]));
        mx1 = fmaxf(mx1, fmaxf(zb.x * INV_TAU + vs[cb],     zb.y * INV_TAU + vs[cb + 1]));
    }
    float mx = fmaxf(mx0, mx1);
    mx = fmaxf(mx, __shfl_xor(mx, 16, 32));     // lanes l and l+16 share a row

    // ---- pass 2: row sum of exp, reduced on the matrix pipe ---------------
#if HAS_WMMA_F32X4
    v8f acc0 = {}; v8f acc1 = {}; v8f acc2 = {}; v8f acc3 = {};
    v2f bones; bones.x = 1.0f; bones.y = 1.0f;  // B = ones(4x16)
    for (int j = 0; j < 128; j += 4) {
        #pragma unroll
        for (int k = 0; k < 4; ++k) {
            float2 z2 = zrow[2 * (j + k) + g];
            int c0 = 4 * (j + k) + 2 * g;
            v2f a;                              // A-layout: VGPR0=K(2g), VGPR1=K(2g+1)
            a.x = __expf(z2.x * INV_TAU + vs[c0]     - mx);
            a.y = __expf(z2.y * INV_TAU + vs[c0 + 1] - mx);
            if (k == 0)      acc0 = __builtin_amdgcn_wmma_f32_16x16x4_f32(false, a, false, bones, (short)0, acc0, false, false);
            else if (k == 1) acc1 = __builtin_amdgcn_wmma_f32_16x16x4_f32(false, a, false, bones, (short)0, acc1, false, false);
            else if (k == 2) acc2 = __builtin_amdgcn_wmma_f32_16x16x4_f32(false, a, false, bones, (short)0, acc2, false, false);
            else             acc3 = __builtin_amdgcn_wmma_f32_16x16x4_f32(false, a, false, bones, (short)0, acc3, false, false);
        }
    }
    v8f acc = acc0 + acc1 + acc2 + acc3;

    // C/D layout: element m on lanes 0-15 is row m, on lanes 16-31 is row m+8.
    const int rowSel = (lane & 7) + ((lane >> 4) << 3);
    float mxw = __shfl(mx, rowSel, 32);         // rowmax of the row we output
    float ssum = 0.0f;
    #pragma unroll
    for (int m = 0; m < 8; ++m) ssum = ((lane & 7) == m) ? acc[m] : ssum;
    if ((lane & 8) == 0) {                      // lanes 0-7 and 16-23: 16 unique rows
        u[(size_t)b * N + row0 + (wv << 4) + rowSel] = -(mxw + __logf(ssum));
    }
#else
    // VALU fallback (keeps kernel correct if the builtin is absent)
    float s = 0.0f;
    for (int j = 0; j < 128; ++j) {
        float2 z2 = zrow[2 * j + g];
        int c0 = 4 * j + 2 * g;
        s += __expf(z2.x * INV_TAU + vs[c0] - mx) + __expf(z2.y * INV_TAU + vs[c0 + 1] - mx);
    }
    s += __shfl_xor(s, 16, 32);
    if (lane < 16) u[(size_t)b * N + row0 + lr] = -(mx + __logf(s));
#endif
}

// ---------------------------------------------------------------------------
// v[b, c] = -logsumexp_r( logits[b,r,c]*INV_TAU + u[b,r] )
// Columns are contiguous in memory -> coalesced running reductions.
// ---------------------------------------------------------------------------
__global__ __launch_bounds__(256)
void sk_col_kernel(const float* __restrict__ logits, const float* __restrict__ u,
                   float* __restrict__ v, int b0)
{
    __shared__ float us[N];
    __shared__ float red[256];

    const int t  = threadIdx.x;
    const int b  = b0 + (blockIdx.x >> 2);
    const int c0 = (blockIdx.x & 3) * 128;
    const int c  = c0 + (t & 127);
    const int h  = t >> 7;                      // row-half
    us[t]       = u[(size_t)b * N + t];
    us[t + 256] = u[(size_t)b * N + t + 256];
    __syncthreads();

    const float* zc  = logits + (size_t)b * N * N + c;
    const int rbeg   = h << 8;

    float m = -3.402823e38f;
    for (int r = 0; r < 256; ++r)
        m = fmaxf(m, zc[(size_t)(rbeg + r) * N] * INV_TAU + us[rbeg + r]);
    red[t] = m;
    __syncthreads();
    const float mc = fmaxf(red[t & 127], red[(t & 127) + 128]);
    __syncthreads();

    float s = 0.0f;
    for (int r = 0; r < 256; ++r)
        s += __expf(zc[(size_t)(rbeg + r) * N] * INV_TAU + us[rbeg + r] - mc);
    red[t] = s;
    __syncthreads();
    if (t < 128) {
        float st = red[t] + red[t + 128];
        v[(size_t)b * N + c0 + t] = -(mc + __logf(st));
    }
}

// ---------------------------------------------------------------------------
// out = exp(Z + u + v), float4 streaming
// ---------------------------------------------------------------------------
__global__ __launch_bounds__(256)
void sk_exp_kernel(const float* __restrict__ logits, const float* __restrict__ u,
                   const float* __restrict__ v, float* __restrict__ out, int b0)
{
    const size_t flat = (size_t)blockIdx.x * 256 + threadIdx.x;  // float4 index
    const int    c4   = (int)(flat & 127);
    const size_t rf   = flat >> 7;
    const int    r    = (int)(rf & 511);
    const int    b    = b0 + (int)(rf >> 9);

    const size_t zoff = (((size_t)b * N) + r) * N + ((size_t)c4 << 2);
    float4 z  = *(const float4*)(logits + zoff);
    float  uu = u[(size_t)b * N + r];
    float4 vv = *(const float4*)(v + (size_t)b * N + ((size_t)c4 << 2));
    float4 o;
    o.x = __expf(z.x * INV_TAU + uu + vv.x);
    o.y = __expf(z.y * INV_TAU + uu + vv.y);
    o.z = __expf(z.z * INV_TAU + uu + vv.z);
    o.w = __expf(z.w * INV_TAU + uu + vv.w);
    *(float4*)(out + zoff) = o;
}

__global__ __launch_bounds__(256)
void sk_zero_kernel(float* __restrict__ v)
{
    v[(size_t)blockIdx.x * 256 + threadIdx.x] = 0.0f;
}

// ---------------------------------------------------------------------------
extern "C" void kernel_launch(void* const* d_in, const int* in_sizes, int n_in,
                              void* d_out, int out_size, void* d_ws, size_t ws_size,
                              hipStream_t stream)
{
    const float* logits = (const float*)d_in[0];
    float* out = (float*)d_out;
    float* u   = (float*)d_ws;                      // [256, 512]
    float* v   = u + (size_t)BATCH * N;             // [256, 512]

    // deterministic init each call (harness does not re-poison between replays)
    sk_zero_kernel<<<dim3((BATCH * N) / 256), dim3(256), 0, stream>>>(v);

    const size_t row_smem = (size_t)(128 * ZS_STRIDE + N) * sizeof(float); // ~260KB of the 320KB LDS

    for (int g = 0; g < NGROUPS; ++g) {
        const int b0 = g * GROUP_B;                 // 64MB group -> L2-resident
        for (int it = 0; it < ITERS; ++it) {
            sk_row_kernel<<<dim3(GROUP_B * 4), dim3(256), row_smem, stream>>>(logits, v, u, b0);
            sk_col_kernel<<<dim3(GROUP_B * 4), dim3(256), 0, stream>>>(logits, u, v, b0);
        }
        sk_exp_kernel<<<dim3((GROUP_B * N * N) / (256 * 4)), dim3(256), 0, stream>>>(logits, u, v, out, b0);
    }
}